// GraphEncoder_22840636080814
// MI455X (gfx1250) — compile-verified
//
#include <hip/hip_runtime.h>

// ---------------------------------------------------------------------------
// CDNA5 (gfx1250) fused graph-encoder.
// Heavy math on V_WMMA_F32_16X16X4_F32 (native fp32 matrix path, wave32).
// ---------------------------------------------------------------------------

typedef __attribute__((ext_vector_type(2))) float v2f;
typedef __attribute__((ext_vector_type(8))) float v8f;

#define HDIM 256

__device__ __forceinline__ v8f wmma_f32_16x16x4(v2f a, v2f b, v8f c) {
    // 8 args: (neg_a, A, neg_b, B, c_mod, C, reuse_a, reuse_b)
    return __builtin_amdgcn_wmma_f32_16x16x4_f32(false, a, false, b,
                                                 (short)0, c, false, false);
}

// ---------------------------------------------------------------------------
// Histogram of graph ids for nodes and edges (also feeds the counting sort).
// ---------------------------------------------------------------------------
__global__ void ge_count_kernel(const int* __restrict__ batch_idx,
                                const int* __restrict__ edge_src,
                                int* __restrict__ ncnt, int* __restrict__ ecnt,
                                int n_nodes, int n_edges) {
    int i = blockIdx.x * blockDim.x + threadIdx.x;
    if (i < n_nodes) atomicAdd(&ncnt[batch_idx[i]], 1);
    if (i < n_edges) atomicAdd(&ecnt[batch_idx[edge_src[i]]], 1);
}

// Exclusive scan over G<=512 bins (single block of 512 threads).
__global__ void ge_scan_kernel(const int* __restrict__ ecnt,
                               int* __restrict__ cursor, int G) {
    __shared__ int s[512];
    int i = threadIdx.x;
    int v = (i < G) ? ecnt[i] : 0;
    s[i] = v;
    __syncthreads();
    for (int off = 1; off < 512; off <<= 1) {
        int t = (i >= off) ? s[i - off] : 0;
        __syncthreads();
        s[i] += t;
        __syncthreads();
    }
    if (i < G) cursor[i] = s[i] - v;   // exclusive prefix
}

// Scatter edge indices into graph-sorted order.
__global__ void ge_scatter_kernel(const int* __restrict__ batch_idx,
                                  const int* __restrict__ edge_src,
                                  int* __restrict__ cursor,
                                  int* __restrict__ perm, int n_edges) {
    int i = blockIdx.x * blockDim.x + threadIdx.x;
    if (i < n_edges) {
        int gid = batch_idx[edge_src[i]];
        int pos = atomicAdd(&cursor[gid], 1);
        perm[pos] = i;
    }
}

// ---------------------------------------------------------------------------
// Fused 2-layer MLP + segment-sum. One wave owns a 16-row tile.
//   layer1 tile (WMMA K-loop) -> ReLU -> per-wave LDS transpose staging
//   -> immediately accumulated into all 16 layer2 N-tiles (acc2 in VGPRs)
//   -> run-length-reduced fp32 atomics into seg_sum[G][H].
// Fragment layouts per ISA 7.12.2:
//   A  (16x4):  lane m, group g holds {A[m][k+2g], A[m][k+2g+1]}
//   B  (4x16):  lane m, group g holds {B[k+2g][m], B[k+2g+1][m]}
//   C/D(16x16): VGPR r, lane -> row (r + 8g), col m
// ---------------------------------------------------------------------------
template <int DIN, bool EDGE>
__global__ __launch_bounds__(256)
void ge_mlp_seg_kernel(const float* __restrict__ X,
                       const float* __restrict__ W1, const float* __restrict__ B1,
                       const float* __restrict__ W2, const float* __restrict__ B2,
                       const int* __restrict__ batch_idx,
                       const int* __restrict__ edge_src,
                       const int* __restrict__ perm,
                       float* __restrict__ seg_sum,
                       int n_rows) {
    __shared__ float hbuf[8 * 256];          // 1KB h-staging per wave
    const int wave = threadIdx.x >> 5;
    const int lane = threadIdx.x & 31;
    const int m = lane & 15;
    const int g = lane >> 4;
    const int rows0 = (blockIdx.x * 8 + wave) * 16;
    if (rows0 >= n_rows) return;             // wave-uniform: EXEC stays all-ones
    float* hw = hbuf + wave * 256;

    // Source row for this lane's A-matrix row (tile row m).
    int arow = rows0 + m;
    if (arow >= n_rows) arow = n_rows - 1;   // padded rows: skipped in epilogue
    if (EDGE && perm) arow = perm[arow];
    const float* xrow = X + (size_t)arow * DIN;

    // Graph ids for the 8 output rows this lane accumulates (M = r + 8g).
    int gid[8];
#pragma unroll
    for (int r = 0; r < 8; ++r) {
        int rr = rows0 + 8 * g + r;
        if (rr < n_rows) {
            int src = (EDGE && perm) ? perm[rr] : rr;
            gid[r] = EDGE ? batch_idx[edge_src[src]] : batch_idx[src];
        } else {
            gid[r] = -1;
        }
    }
    bool uni = true;
#pragma unroll
    for (int r = 1; r < 8; ++r) uni = uni && (gid[r] == gid[0]);
    uni = uni && (gid[0] >= 0);

    // Layer-2 accumulators: 16 N-tiles, bias-initialized.
    v8f acc2[16];
#pragma unroll
    for (int n2 = 0; n2 < 16; ++n2) {
        float b = B2[n2 * 16 + m];
#pragma unroll
        for (int r = 0; r < 8; ++r) acc2[n2][r] = b;
    }

#pragma unroll 1
    for (int n1 = 0; n1 < 16; ++n1) {
        float b1v = B1[n1 * 16 + m];
        v8f acc1;
#pragma unroll
        for (int r = 0; r < 8; ++r) acc1[r] = b1v;

        // h[:,16*n1:16*n1+16] = x @ W1 tile
#pragma unroll 4
        for (int k = 0; k < DIN; k += 4) {
            v2f a = *(const v2f*)(xrow + k + 2 * g);
            v2f b;
            b.x = W1[(size_t)(k + 2 * g) * HDIM + n1 * 16 + m];
            b.y = W1[(size_t)(k + 2 * g + 1) * HDIM + n1 * 16 + m];
            acc1 = wmma_f32_16x16x4(a, b, acc1);
        }

        // ReLU + stage to per-wave LDS (row-major 16x16) for the D->A transpose.
#pragma unroll
        for (int r = 0; r < 8; ++r) {
            float v = acc1[r];
            hw[(r + 8 * g) * 16 + m] = v > 0.f ? v : 0.f;
        }
        asm volatile("s_wait_dscnt 0" ::: "memory");

        // Fold this K-chunk of 16 into every layer-2 N-tile.
#pragma unroll 1
        for (int k2 = 0; k2 < 4; ++k2) {
            v2f a2 = *(const v2f*)(hw + m * 16 + k2 * 4 + 2 * g);
            const float* w2p = W2 + (size_t)(n1 * 16 + k2 * 4 + 2 * g) * HDIM + m;
#pragma unroll
            for (int n2 = 0; n2 < 16; ++n2) {
                v2f b2;
                b2.x = w2p[n2 * 16];
                b2.y = w2p[HDIM + n2 * 16];
                acc2[n2] = wmma_f32_16x16x4(a2, b2, acc2[n2]);
            }
        }
    }

    // Segment-sum epilogue with in-register run-length reduction
    // (graph-sorted rows -> mostly 1 atomic per 8 rows).
#pragma unroll 1
    for (int n2 = 0; n2 < 16; ++n2) {
        int col = n2 * 16 + m;
        if (uni) {
            float s = 0.f;
#pragma unroll
            for (int r = 0; r < 8; ++r) s += acc2[n2][r];
            atomicAdd(&seg_sum[(size_t)gid[0] * HDIM + col], s);
        } else {
#pragma unroll
            for (int r = 0; r < 8; ++r)
                if (gid[r] >= 0)
                    atomicAdd(&seg_sum[(size_t)gid[r] * HDIM + col], acc2[n2][r]);
        }
    }
}

// ---------------------------------------------------------------------------
// Finalize: out = (node_sum/ncnt + edge_sum/ecnt) @ Wc + bc  (guarded means),
// mean fused into the WMMA A-fragment load. One wave per 16 graph rows.
// ---------------------------------------------------------------------------
__global__ __launch_bounds__(256)
void ge_finalize_kernel(const float* __restrict__ node_sum,
                        const float* __restrict__ edge_sum,
                        const int* __restrict__ ncnt, const int* __restrict__ ecnt,
                        const float* __restrict__ Wc, const float* __restrict__ bc,
                        float* __restrict__ out, int G) {
    const int wave = threadIdx.x >> 5;
    const int lane = threadIdx.x & 31;
    const int m = lane & 15;
    const int g = lane >> 4;
    const int rows0 = (blockIdx.x * 8 + wave) * 16;
    if (rows0 >= G) return;

    const int row = rows0 + m;
    const int nc = ncnt[row], ec = ecnt[row];
    const float rn = nc > 0 ? 1.f / (float)nc : 0.f;
    const float re = ec > 0 ? 1.f / (float)ec : 0.f;
    const float* ns = node_sum + (size_t)row * HDIM;
    const float* es = edge_sum + (size_t)row * HDIM;

    v8f acc[16];
#pragma unroll
    for (int n2 = 0; n2 < 16; ++n2) {
        float b = bc[n2 * 16 + m];
#pragma unroll
        for (int r = 0; r < 8; ++r) acc[n2][r] = b;
    }

#pragma unroll 1
    for (int k = 0; k < HDIM; k += 4) {
        v2f nsv = *(const v2f*)(ns + k + 2 * g);
        v2f esv = *(const v2f*)(es + k + 2 * g);
        v2f a;
        a.x = nsv.x * rn + esv.x * re;
        a.y = nsv.y * rn + esv.y * re;
        const float* wp = Wc + (size_t)(k + 2 * g) * HDIM + m;
#pragma unroll
        for (int n2 = 0; n2 < 16; ++n2) {
            v2f b;
            b.x = wp[n2 * 16];
            b.y = wp[HDIM + n2 * 16];
            acc[n2] = wmma_f32_16x16x4(a, b, acc[n2]);
        }
    }

#pragma unroll
    for (int n2 = 0; n2 < 16; ++n2)
#pragma unroll
        for (int r = 0; r < 8; ++r)
            out[(size_t)(rows0 + r + 8 * g) * HDIM + n2 * 16 + m] = acc[n2][r];
}

// ---------------------------------------------------------------------------
// Host launcher (graph-capture safe: only async ops on `stream`).
// ---------------------------------------------------------------------------
extern "C" void kernel_launch(void* const* d_in, const int* in_sizes, int n_in,
                              void* d_out, int out_size, void* d_ws, size_t ws_size,
                              hipStream_t stream) {
    const float* x         = (const float*)d_in[0];
    const float* edge_attr = (const float*)d_in[1];
    const int*   batch_idx = (const int*)d_in[2];
    const int*   edge_src  = (const int*)d_in[3];
    const float* Wn1 = (const float*)d_in[4];
    const float* bn1 = (const float*)d_in[5];
    const float* Wn2 = (const float*)d_in[6];
    const float* bn2 = (const float*)d_in[7];
    const float* We1 = (const float*)d_in[8];
    const float* be1 = (const float*)d_in[9];
    const float* We2 = (const float*)d_in[10];
    const float* be2 = (const float*)d_in[11];
    const float* Wc  = (const float*)d_in[12];
    const float* bc  = (const float*)d_in[13];

    const int D_NODE = 128, D_EDGE = 64, G = 512;
    const int N = in_sizes[0] / D_NODE;
    const int E = in_sizes[1] / D_EDGE;

    // Workspace layout.
    char* ws = (char*)d_ws;
    float* node_sum = (float*)ws;                                   // G*H f32
    float* edge_sum = (float*)(ws + (size_t)G * HDIM * 4);          // G*H f32
    int*   ncnt     = (int*)(ws + (size_t)2 * G * HDIM * 4);        // G
    int*   ecnt     = ncnt + G;                                     // G
    int*   cursor   = ncnt + 2 * G;                                 // G
    int*   perm     = ncnt + 3 * G;                                 // E
    size_t base_bytes = (size_t)2 * G * HDIM * 4 + (size_t)3 * G * 4;
    size_t sort_bytes = base_bytes + (size_t)E * 4;
    bool do_sort = ws_size >= sort_bytes;

    // Zero sums + histograms (+cursor) every call.
    hipMemsetAsync(d_ws, 0, base_bytes, stream);

    {
        int n = (N > E) ? N : E;
        ge_count_kernel<<<(n + 255) / 256, 256, 0, stream>>>(
            batch_idx, edge_src, ncnt, ecnt, N, E);
    }

    const int* permp = nullptr;
    if (do_sort) {
        ge_scan_kernel<<<1, 512, 0, stream>>>(ecnt, cursor, G);
        ge_scatter_kernel<<<(E + 255) / 256, 256, 0, stream>>>(
            batch_idx, edge_src, cursor, perm, E);
        permp = perm;
    }

    {
        int tiles = (N + 15) / 16;
        ge_mlp_seg_kernel<128, false><<<(tiles + 7) / 8, 256, 0, stream>>>(
            x, Wn1, bn1, Wn2, bn2, batch_idx, edge_src, nullptr, node_sum, N);
    }
    {
        int tiles = (E + 15) / 16;
        ge_mlp_seg_kernel<64, true><<<(tiles + 7) / 8, 256, 0, stream>>>(
            edge_attr, We1, be1, We2, be2, batch_idx, edge_src, permp, edge_sum, E);
    }

    {
        int tiles = (G + 15) / 16;
        ge_finalize_kernel<<<(tiles + 7) / 8, 256, 0, stream>>>(
            node_sum, edge_sum, ncnt, ecnt, Wc, bc, (float*)d_out, G);
    }
}